// s2net_80805514707354
// MI455X (gfx1250) — compile-verified
//
#include <hip/hip_runtime.h>
#include <math.h>

// ---------------- problem constants ----------------
#define BATCH 128
#define NBIN  15          // B_IN
#define NB1   10          // B1
#define NB2   6           // B2
#define NF1   20          // F1
#define NF2   40          // F2
#define NFOUT 10          // F_OUT
#define M1    19          // 2*B1-1
#define M2    11          // 2*B2-1
#define NBETAIN 30        // 2*B_IN
#define NA1   20          // 2*B1
#define NA2   12          // 2*B2

// Z2 GEMM dims (per l): Z2[(b,m),(o,n)] = sum_{(i,k)} X2A * conj(psi2B)
#define MR 1408           // BATCH*M2
#define KR 220            // NF1*M2
#define NR 440            // NF2*M2   (valid cols)
#define NRP 448           // padded col stride (28 x 16-tiles)
#define BSO3_W 1452       // valid cols of B_SO3 as real matrix
#define BSO3_WP 1456      // padded stride (91 x 16-tiles)

#define PI_D 3.14159265358979323846
#define PI_F 3.14159265358979323846f

// ---------------- workspace layout (float offsets) ----------------
constexpr size_t OFF_WS2F  = 0;                          // [30][10][19]            5700
constexpr size_t OFF_WINV1 = 5700;                       // [20][10][19][19]        72200
constexpr size_t OFF_WSO3F = 77900;                      // [20][6][11][11]         14520
constexpr size_t OFF_WINV2 = 92420;                      // [12][6][11][11]         8712
constexpr size_t OFF_BS2   = 101132;                     // [24][10][19] cplx       9120
constexpr size_t OFF_BSO3  = 110252;                     // [144][1456] (padded)    209664
constexpr size_t OFF_EA1   = 319916;                     // [19][20] cplx           760
constexpr size_t OFF_EA2   = 320676;                     // [11][12] cplx           264
constexpr size_t OFF_WINT  = 320940;                     // [12]                    12
constexpr size_t OFF_XM    = 320952;                     // [128][30][19] cplx      145920
constexpr size_t OFF_X     = 466872;                     // [128][10][19] cplx      48640
constexpr size_t OFF_PSI   = 515512;                     // [20][10][19] cplx       7600
constexpr size_t OFF_X2A   = 523112;                     // [6][2][1408][220]       3717120
constexpr size_t OFF_PSI2B = 4240232;                    // [6][2][220][448]        1182720
constexpr size_t OFF_Z2    = 5422952;                    // [6][2][1408][448]       7569408
constexpr size_t OFF_FEAT  = 12992360;                   // [128][40]               5120

// ---------------- WMMA ----------------
typedef float v2f __attribute__((ext_vector_type(2)));
typedef float v8f __attribute__((ext_vector_type(8)));

static __device__ __forceinline__ v8f wmma4(v2f a, v2f b, v8f c) {
  // V_WMMA_F32_16X16X4_F32
  return __builtin_amdgcn_wmma_f32_16x16x4_f32(false, a, false, b, (short)0, c, false, false);
}

// ---------------- double-precision Wigner helpers ----------------
static __device__ __forceinline__ double dfact(int k) {
  double f = 1.0;
  for (int i = 2; i <= k; ++i) f *= (double)i;
  return f;
}
static __device__ __forceinline__ double dipow(double x, int e) {
  double r = 1.0;
  for (int i = 0; i < e; ++i) r *= x;
  return r;
}
static __device__ double wig_d(int l, int m, int n, double beta) {
  double cb = cos(0.5 * beta), sb = sin(0.5 * beta);
  double pref = sqrt(dfact(l + m) * dfact(l - m) * dfact(l + n) * dfact(l - n));
  int s0 = (n - m > 0) ? (n - m) : 0;
  int s1 = (l + n < l - m) ? (l + n) : (l - m);
  double v = 0.0;
  for (int s = s0; s <= s1; ++s) {
    double term = dipow(cb, 2 * l + n - m - 2 * s) * dipow(sb, m - n + 2 * s) /
                  (dfact(l + n - s) * dfact(s) * dfact(m - n + s) * dfact(l - m - s));
    v += ((m - n + s) & 1) ? -term : term;
  }
  return pref * v;
}
static __device__ double quad_w(int bq, int k) {
  double beta = PI_D * (2 * k + 1) / (4.0 * bq);
  double s = 0.0;
  for (int j = 0; j < bq; ++j) s += sin((2 * j + 1) * beta) / (double)(2 * j + 1);
  return 2.0 / bq * sin(beta) * s;
}

// ============== K0: regenerate all module constants on device ==============
__global__ __launch_bounds__(256) void s2net_const_k(float* __restrict__ ws) {
  long idx = (long)blockIdx.x * 256 + threadIdx.x;
  if (idx < 5700) {                                   // W_S2_FWD [k<30][l<10][mi<19]
    int k = idx / 190, r = idx % 190, l = r / 19, m = r % 19 - 9;
    float v = 0.f;
    if (m >= -l && m <= l) {
      double beta = PI_D * (2 * k + 1) / (4.0 * NBIN);
      v = (float)(wig_d(l, m, 0, beta) * quad_w(NBIN, k));
    }
    ws[OFF_WS2F + idx] = v; return;
  }
  idx -= 5700;
  if (idx < 72200) {                                  // W_INV1 [k<20][l<10][mi][ni]
    int k = idx / 3610, r = idx % 3610, l = r / 361;
    int m = (r % 361) / 19 - 9, n = r % 19 - 9;
    float v = 0.f;
    if (abs(m) <= l && abs(n) <= l) {
      double beta = PI_D * (2 * k + 1) / (4.0 * NB1);
      v = (float)(wig_d(l, m, n, beta) * (double)(2 * l + 1));
    }
    ws[OFF_WINV1 + idx] = v; return;
  }
  idx -= 72200;
  if (idx < 14520) {                                  // W_SO3_FWD [k<20][l<6][mi][ni]
    int k = idx / 726, r = idx % 726, l = r / 121;
    int m = (r % 121) / 11 - 5, n = r % 11 - 5;
    float v = 0.f;
    if (abs(m) <= l && abs(n) <= l) {
      double beta = PI_D * (2 * k + 1) / (4.0 * NB1);
      v = (float)(wig_d(l, m, n, beta) * quad_w(NB1, k));
    }
    ws[OFF_WSO3F + idx] = v; return;
  }
  idx -= 14520;
  if (idx < 8712) {                                   // W_INV2 [k<12][l<6][mi][ni]
    int k = idx / 726, r = idx % 726, l = r / 121;
    int m = (r % 121) / 11 - 5, n = r % 11 - 5;
    float v = 0.f;
    if (abs(m) <= l && abs(n) <= l) {
      double beta = PI_D * (2 * k + 1) / (4.0 * NB2);
      v = (float)(wig_d(l, m, n, beta) * (double)(2 * l + 1));
    }
    ws[OFF_WINV2 + idx] = v; return;
  }
  idx -= 8712;
  if (idx < 4560) {                                   // B_S2 [p<24][l<10][mi] complex
    int p = idx / 190, r = idx % 190, l = r / 19, m = r % 19 - 9;
    float re = 0.f, im = 0.f;
    if (abs(m) <= l) {
      double beta = (double)(p / 8 + 1) * (PI_D / 24.0);
      double alpha = 2.0 * PI_D * (double)(p % 8) / 8.0;
      double d = wig_d(l, m, 0, beta);
      double sn, cs; sincos(-(double)m * alpha, &sn, &cs);
      re = (float)(d * cs); im = (float)(d * sn);
    }
    ws[OFF_BS2 + 2 * idx] = re; ws[OFF_BS2 + 2 * idx + 1] = im; return;
  }
  idx -= 4560;
  if (idx < 104544) {                                 // B_SO3 [p<144][726 cplx], stride 1456
    int p = idx / 726, r = idx % 726, l = r / 121;
    int m = (r % 121) / 11 - 5, n = r % 11 - 5;
    float re = 0.f, im = 0.f;
    if (abs(m) <= l && abs(n) <= l) {
      int bi = p / 48, ai = (p / 6) % 8, gi = p % 6;
      double beta = (double)(bi + 1) * (PI_D / 24.0);
      double alpha = 2.0 * PI_D * (double)ai / 8.0;
      double gamma = 2.0 * PI_D * (double)gi / 6.0;
      double d = wig_d(l, m, n, beta);
      double sn, cs; sincos(-((double)m * alpha + (double)n * gamma), &sn, &cs);
      re = (float)(d * cs); im = (float)(d * sn);
    }
    ws[OFF_BSO3 + (size_t)p * BSO3_WP + 2 * r] = re;
    ws[OFF_BSO3 + (size_t)p * BSO3_WP + 2 * r + 1] = im; return;
  }
  idx -= 104544;
  if (idx < 576) {                                    // B_SO3 pad cols [1452..1455] = 0
    int p = idx / 4, q = idx % 4;
    ws[OFF_BSO3 + (size_t)p * BSO3_WP + BSO3_W + q] = 0.f; return;
  }
  idx -= 576;
  if (idx < 380) {                                    // E_A1 [mi<19][a<20]
    int mi = idx / 20, a = idx % 20;
    double sn, cs; sincos((double)(mi - 9) * 2.0 * PI_D * (double)a / 20.0, &sn, &cs);
    ws[OFF_EA1 + 2 * idx] = (float)cs; ws[OFF_EA1 + 2 * idx + 1] = (float)sn; return;
  }
  idx -= 380;
  if (idx < 132) {                                    // E_A2 [mi<11][a<12]
    int mi = idx / 12, a = idx % 12;
    double sn, cs; sincos((double)(mi - 5) * 2.0 * PI_D * (double)a / 12.0, &sn, &cs);
    ws[OFF_EA2 + 2 * idx] = (float)cs; ws[OFF_EA2 + 2 * idx + 1] = (float)sn; return;
  }
  idx -= 132;
  if (idx < 12) ws[OFF_WINT + idx] = (float)quad_w(NB2, (int)idx);
}

// ============== K1a: last-axis DFT (30 -> 19 freqs) ==============
__global__ __launch_bounds__(256) void s2net_fft1_k(const float* __restrict__ x,
                                                    float* __restrict__ ws) {
  int idx = blockIdx.x * 256 + threadIdx.x;
  if (idx >= BATCH * NBETAIN * M1) return;
  int b = idx / (NBETAIN * M1), r = idx % (NBETAIN * M1);
  int k = r / M1, m = r % M1 - 9;
  const float* xp = x + ((size_t)b * NBETAIN + k) * NBETAIN;
  float ar = 0.f, ai = 0.f;
  for (int j = 0; j < NBETAIN; ++j) {
    int t0 = ((m * j) % 30 + 30) % 30;
    float sn, cs; sincosf(-2.0f * PI_F * (float)t0 / 30.0f, &sn, &cs);
    ar += xp[j] * cs; ai += xp[j] * sn;
  }
  ((float2*)(ws + OFF_XM))[idx] = make_float2(ar, ai);
}

// ============== K1b: X[b,l,m] = sum_k W_S2_FWD[k,l,m] * xm[b,k,m] ==============
__global__ __launch_bounds__(256) void s2net_x_k(float* __restrict__ ws) {
  int idx = blockIdx.x * 256 + threadIdx.x;
  if (idx >= BATCH * NB1 * M1) return;
  int b = idx / (NB1 * M1), lm = idx % (NB1 * M1);
  const float2* xm = (const float2*)(ws + OFF_XM);
  float ar = 0.f, ai = 0.f;
  for (int k = 0; k < NBETAIN; ++k) {
    float w = ws[OFF_WS2F + (size_t)k * (NB1 * M1) + lm];
    float2 v = xm[((size_t)b * NBETAIN + k) * M1 + (lm % M1)];
    ar += w * v.x; ai += w * v.y;
  }
  ((float2*)(ws + OFF_X))[idx] = make_float2(ar, ai);
}

// ============== K2: psi[o,l,m] = sum_p kernel1[o,p]*B_S2[p,l,m] ==============
__global__ __launch_bounds__(256) void s2net_psi_k(const float* __restrict__ k1,
                                                   float* __restrict__ ws) {
  int idx = blockIdx.x * 256 + threadIdx.x;
  if (idx >= NF1 * NB1 * M1) return;
  int o = idx / (NB1 * M1), lm = idx % (NB1 * M1);
  const float2* bs2 = (const float2*)(ws + OFF_BS2);
  float ar = 0.f, ai = 0.f;
  for (int p = 0; p < 24; ++p) {
    float w = k1[o * 24 + p];
    float2 v = bs2[(size_t)p * (NB1 * M1) + lm];
    ar += w * v.x; ai += w * v.y;
  }
  ((float2*)(ws + OFF_PSI))[idx] = make_float2(ar, ai);
}

// ============== K3: fused Z -> W_INV1 -> iDFT2 -> relu -> DFT2 -> W_SO3_FWD ==============
__global__ __launch_bounds__(256) void s2net_mid_k(float* __restrict__ ws) {
  const int b = blockIdx.x, i = blockIdx.y, tid = threadIdx.x;
  const float2* Xc = (const float2*)(ws + OFF_X);
  const float2* Pc = (const float2*)(ws + OFF_PSI);
  const float* Winv1 = ws + OFF_WINV1;
  const float* Wso3 = ws + OFF_WSO3F;
  const float2* Ea = (const float2*)(ws + OFF_EA1);
  float* X2A = ws + OFF_X2A;

  __shared__ float2 sX[NB1][M1];
  __shared__ float2 sP[NB1][M1];
  __shared__ float2 sE[M1][NA1];
  __shared__ float2 sfh[M1][M1];
  __shared__ float2 sT[M1][NA1];
  __shared__ float  sY[NA1][NA1];
  __shared__ float2 sQ[NA1][M2];

  for (int t = tid; t < NB1 * M1; t += 256) {
    sX[t / M1][t % M1] = Xc[(size_t)b * NB1 * M1 + t];
    sP[t / M1][t % M1] = Pc[(size_t)i * NB1 * M1 + t];
  }
  for (int t = tid; t < M1 * NA1; t += 256) sE[t / NA1][t % NA1] = Ea[t];
  float2 x2a[NB2];
  for (int l = 0; l < NB2; ++l) x2a[l] = make_float2(0.f, 0.f);
  __syncthreads();

  for (int k = 0; k < NA1; ++k) {
    for (int t = tid; t < M1 * M1; t += 256) {
      int mi = t / M1, ni = t % M1;
      float ar = 0.f, ai = 0.f;
      const float* wp = Winv1 + ((size_t)k * NB1 * M1 + mi) * M1 + ni;
      for (int l = 0; l < NB1; ++l) {
        float w = wp[(size_t)l * M1 * M1];
        float2 xv = sX[l][mi], pv = sP[l][ni];
        float zr = xv.x * pv.x + xv.y * pv.y;
        float zi = xv.y * pv.x - xv.x * pv.y;
        ar += w * zr; ai += w * zi;
      }
      sfh[mi][ni] = make_float2(ar, ai);
    }
    __syncthreads();
    for (int t = tid; t < M1 * NA1; t += 256) {
      int m = t / NA1, g = t % NA1;
      float ar = 0.f, ai = 0.f;
      for (int n = 0; n < M1; ++n) {
        float2 f = sfh[m][n], e = sE[n][g];
        ar += f.x * e.x - f.y * e.y;
        ai += f.x * e.y + f.y * e.x;
      }
      sT[m][g] = make_float2(ar, ai);
    }
    __syncthreads();
    for (int t = tid; t < NA1 * NA1; t += 256) {
      int a = t / NA1, g = t % NA1;
      float acc = 0.f;
      for (int m = 0; m < M1; ++m) {
        float2 e = sE[m][a], tv = sT[m][g];
        acc += e.x * tv.x - e.y * tv.y;
      }
      sY[a][g] = fmaxf(acc, 0.f);
    }
    __syncthreads();
    for (int t = tid; t < NA1 * M2; t += 256) {
      int a = t / M2, n2 = t % M2;
      float ar = 0.f, ai = 0.f;
      for (int g = 0; g < NA1; ++g) {
        float yv = sY[a][g]; float2 e = sE[n2 + 4][g];
        ar += yv * e.x; ai -= yv * e.y;
      }
      sQ[a][n2] = make_float2(ar, ai);
    }
    __syncthreads();
    if (tid < M2 * M2) {
      int m2 = tid / M2, n2 = tid % M2;
      float ar = 0.f, ai = 0.f;
      for (int a = 0; a < NA1; ++a) {
        float2 e = sE[m2 + 4][a], q = sQ[a][n2];
        ar += e.x * q.x + e.y * q.y;
        ai += e.x * q.y - e.y * q.x;
      }
      for (int l = 0; l < NB2; ++l) {
        float w = Wso3[((size_t)(k * NB2 + l) * M2 + m2) * M2 + n2];
        x2a[l].x += w * ar; x2a[l].y += w * ai;
      }
    }
    __syncthreads();
  }
  if (tid < M2 * M2) {
    int m2 = tid / M2, n2 = tid % M2;
    int row = b * M2 + m2, col = i * M2 + n2;
    for (int l = 0; l < NB2; ++l) {
      X2A[((size_t)(l * 2 + 0) * MR + row) * KR + col] = x2a[l].x;
      X2A[((size_t)(l * 2 + 1) * MR + row) * KR + col] = x2a[l].y;
    }
  }
}

// ============== K4: psi2 via WMMA real GEMM (800 x 144 x 1456 padded), scatter ==============
__global__ __launch_bounds__(256) void s2net_psi2_k(const float* __restrict__ k2,
                                                    float* __restrict__ ws) {
  int wv = (blockIdx.x * 256 + (int)threadIdx.x) >> 5;
  if (wv >= 50 * 91) return;
  int lane = threadIdx.x & 31, half = lane >> 4, lm = lane & 15;
  int mt = wv / 91, nt = wv % 91;
  const float* Bm = ws + OFF_BSO3;                    // [144][1456], pad cols zeroed
  float* P2 = ws + OFF_PSI2B;                         // [6][2][220][448]
  v8f acc = {0.f, 0.f, 0.f, 0.f, 0.f, 0.f, 0.f, 0.f};
  int arow = mt * 16 + lm;
  int bcol = nt * 16 + lm;
  for (int k0 = 0; k0 < 144; k0 += 4) {
    int kk = k0 + 2 * half;
    v2f a = {k2[(size_t)arow * 144 + kk], k2[(size_t)arow * 144 + kk + 1]};
    v2f bb = {Bm[(size_t)kk * BSO3_WP + bcol], Bm[(size_t)(kk + 1) * BSO3_WP + bcol]};
    acc = wmma4(a, bb, acc);
  }
  if (bcol < BSO3_W) {
    int l = bcol / 242, r1 = bcol % 242;
    int mp = r1 / 22, r2 = r1 % 22, np_ = r2 / 2, ri = r2 & 1;
    for (int r = 0; r < 8; ++r) {
      int rho = mt * 16 + r + 8 * half;               // rho = i*40 + o
      int ii = rho / NF2, oo = rho % NF2;
      P2[((size_t)(l * 2 + ri) * KR + ii * M2 + np_) * NRP + oo * M2 + mp] = acc[r];
    }
  }
}

// ============== K4b: zero psi2B pad columns [440..447] ==============
__global__ __launch_bounds__(256) void s2net_padb_k(float* __restrict__ ws) {
  int idx = blockIdx.x * 256 + threadIdx.x;
  if (idx >= 12 * KR * 8) return;
  int lr = idx / (KR * 8), r = idx % (KR * 8), row = r / 8, c = r % 8;
  ws[OFF_PSI2B + (size_t)lr * KR * NRP + (size_t)row * NRP + NR + c] = 0.f;
}

// ============== K5: Z2 complex GEMM, 16x64 strip per wave, branchless ==============
__global__ __launch_bounds__(256) void s2net_z2_k(float* __restrict__ ws) {
  int wv = (blockIdx.x * 256 + (int)threadIdx.x) >> 5;
  int lane = threadIdx.x & 31, half = lane >> 4, lm = lane & 15;
  int l = wv / 616, rem = wv % 616, mt = rem / 7, ntg = rem % 7;
  const float* Ar = ws + OFF_X2A + (size_t)(l * 2 + 0) * MR * KR + (size_t)(mt * 16 + lm) * KR;
  const float* Ai = ws + OFF_X2A + (size_t)(l * 2 + 1) * MR * KR + (size_t)(mt * 16 + lm) * KR;
  const float* Br = ws + OFF_PSI2B + (size_t)(l * 2 + 0) * KR * NRP + ntg * 64 + lm;
  const float* Bi = ws + OFF_PSI2B + (size_t)(l * 2 + 1) * KR * NRP + ntg * 64 + lm;
  v8f cre[4], cim[4];
#pragma unroll
  for (int t = 0; t < 4; ++t) {
    cre[t] = (v8f){0.f, 0.f, 0.f, 0.f, 0.f, 0.f, 0.f, 0.f};
    cim[t] = (v8f){0.f, 0.f, 0.f, 0.f, 0.f, 0.f, 0.f, 0.f};
  }
  for (int k0 = 0; k0 < KR; k0 += 4) {
    int kk = k0 + 2 * half;
    __builtin_prefetch(Ar + kk + 64, 0, 3);
    v2f ar = {Ar[kk], Ar[kk + 1]};
    v2f ai = {Ai[kk], Ai[kk + 1]};
    v2f arn = -ar;
#pragma unroll
    for (int t = 0; t < 4; ++t) {
      v2f br = {Br[(size_t)kk * NRP + t * 16], Br[(size_t)(kk + 1) * NRP + t * 16]};
      v2f bi = {Bi[(size_t)kk * NRP + t * 16], Bi[(size_t)(kk + 1) * NRP + t * 16]};
      // Z = A * conj(B): Re = Ar*Br + Ai*Bi ; Im = Ai*Br - Ar*Bi
      cre[t] = wmma4(ar, br, cre[t]);
      cre[t] = wmma4(ai, bi, cre[t]);
      cim[t] = wmma4(ai, br, cim[t]);
      cim[t] = wmma4(arn, bi, cim[t]);
    }
  }
  float* Zr = ws + OFF_Z2 + (size_t)(l * 2 + 0) * MR * NRP;
  float* Zi = ws + OFF_Z2 + (size_t)(l * 2 + 1) * MR * NRP;
#pragma unroll
  for (int t = 0; t < 4; ++t) {
    int col = ntg * 64 + t * 16 + lm;
#pragma unroll
    for (int r = 0; r < 8; ++r) {
      int row = mt * 16 + r + 8 * half;
      Zr[(size_t)row * NRP + col] = cre[t][r];
      Zi[(size_t)row * NRP + col] = cim[t][r];
    }
  }
}

// ============== K6: fused W_INV2 -> iDFT2 -> relu -> quad integral ==============
__global__ __launch_bounds__(256) void s2net_fin_k(float* __restrict__ ws) {
  const int b = blockIdx.x, o = blockIdx.y, tid = threadIdx.x;
  const float* Z = ws + OFF_Z2;
  const float* Winv2 = ws + OFF_WINV2;
  const float2* Ea = (const float2*)(ws + OFF_EA2);
  const float* wint = ws + OFF_WINT;
  __shared__ float2 sfh[NA2][M2][M2];
  __shared__ float2 sE[M2][NA2];
  __shared__ float2 sT[M2][NA2];
  __shared__ float sred[256];
  for (int t = tid; t < M2 * NA2; t += 256) sE[t / NA2][t % NA2] = Ea[t];
  for (int t = tid; t < NA2 * M2 * M2; t += 256) {
    int k = t / 121, r = t % 121, m = r / 11, n = r % 11;
    int row = b * M2 + m, col = o * M2 + n;
    float ar = 0.f, ai = 0.f;
    for (int l = 0; l < NB2; ++l) {
      float w = Winv2[((size_t)(k * NB2 + l) * M2 + m) * M2 + n];
      float zr = Z[((size_t)(l * 2 + 0) * MR + row) * NRP + col];
      float zi = Z[((size_t)(l * 2 + 1) * MR + row) * NRP + col];
      ar += w * zr; ai += w * zi;
    }
    sfh[k][m][n] = make_float2(ar, ai);
  }
  __syncthreads();
  float part = 0.f;
  for (int k = 0; k < NA2; ++k) {
    for (int t = tid; t < M2 * NA2; t += 256) {
      int m = t / NA2, g = t % NA2;
      float ar = 0.f, ai = 0.f;
      for (int n = 0; n < M2; ++n) {
        float2 f = sfh[k][m][n], e = sE[n][g];
        ar += f.x * e.x - f.y * e.y;
        ai += f.x * e.y + f.y * e.x;
      }
      sT[m][g] = make_float2(ar, ai);
    }
    __syncthreads();
    float wk = wint[k];
    for (int t = tid; t < NA2 * NA2; t += 256) {
      int a = t / NA2, g = t % NA2;
      float acc = 0.f;
      for (int m = 0; m < M2; ++m) {
        float2 e = sE[m][a], tv = sT[m][g];
        acc += e.x * tv.x - e.y * tv.y;
      }
      part += fmaxf(acc, 0.f) * wk;
    }
    __syncthreads();
  }
  sred[tid] = part;
  __syncthreads();
  for (int s = 128; s > 0; s >>= 1) {
    if (tid < s) sred[tid] += sred[tid + s];
    __syncthreads();
  }
  if (tid == 0) ws[OFF_FEAT + (size_t)b * NF2 + o] = sred[0] * (1.0f / 144.0f);
}

// ============== K7: final linear layer ==============
__global__ __launch_bounds__(256) void s2net_lin_k(const float* __restrict__ ws,
                                                   const float* __restrict__ wlin,
                                                   const float* __restrict__ blin,
                                                   float* __restrict__ out) {
  int idx = blockIdx.x * 256 + threadIdx.x;
  if (idx >= BATCH * NFOUT) return;
  int b = idx / NFOUT, f = idx % NFOUT;
  float acc = blin[f];
  const float* fp = ws + OFF_FEAT + (size_t)b * NF2;
  for (int o = 0; o < NF2; ++o) acc += fp[o] * wlin[f * NF2 + o];
  out[idx] = acc;
}

// ---------------- host ----------------
extern "C" void kernel_launch(void* const* d_in, const int* in_sizes, int n_in,
                              void* d_out, int out_size, void* d_ws, size_t ws_size,
                              hipStream_t stream) {
  const float* x    = (const float*)d_in[0];
  const float* k1   = (const float*)d_in[1];
  const float* k2   = (const float*)d_in[2];
  const float* wlin = (const float*)d_in[3];
  const float* blin = (const float*)d_in[4];
  float* ws  = (float*)d_ws;
  float* out = (float*)d_out;

  s2net_const_k<<<826, 256, 0, stream>>>(ws);
  s2net_fft1_k<<<(BATCH * NBETAIN * M1 + 255) / 256, 256, 0, stream>>>(x, ws);
  s2net_x_k<<<(BATCH * NB1 * M1 + 255) / 256, 256, 0, stream>>>(ws);
  s2net_psi_k<<<(NF1 * NB1 * M1 + 255) / 256, 256, 0, stream>>>(k1, ws);
  dim3 g3(BATCH, NF1);
  s2net_mid_k<<<g3, 256, 0, stream>>>(ws);
  s2net_psi2_k<<<(50 * 91 + 7) / 8, 256, 0, stream>>>(k2, ws);
  s2net_padb_k<<<(12 * KR * 8 + 255) / 256, 256, 0, stream>>>(ws);
  s2net_z2_k<<<(6 * 88 * 7) / 8, 256, 0, stream>>>(ws);
  dim3 g6(BATCH, NF2);
  s2net_fin_k<<<g6, 256, 0, stream>>>(ws);
  s2net_lin_k<<<(BATCH * NFOUT + 255) / 256, 256, 0, stream>>>(ws, wlin, blin, out);
}